// TimeMix_13340168421847
// MI455X (gfx1250) — compile-verified
//
#include <hip/hip_runtime.h>

typedef float v2f __attribute__((ext_vector_type(2)));
typedef float v4f __attribute__((ext_vector_type(4)));
typedef float v8f __attribute__((ext_vector_type(8)));

#define DSZ   4096
#define TILE  16
#define BLOCK 256
#define WAVES 8            // BLOCK / 32
#define KSEG  (DSZ / WAVES) // 512 K-values per wave

// ---------------------------------------------------------------------------
// Kernel 0: token mixing  xk/xv/xr = x*m + state*(1-m)
// ---------------------------------------------------------------------------
__global__ __launch_bounds__(BLOCK) void mix_kernel(
    const float* __restrict__ x, const float* __restrict__ sx,
    const float* __restrict__ tmk, const float* __restrict__ tmv,
    const float* __restrict__ tmr,
    float* __restrict__ xk, float* __restrict__ xv, float* __restrict__ xr)
{
  int i = blockIdx.x * blockDim.x + threadIdx.x;
  if (i < DSZ) {
    float xi = x[i], si = sx[i];
    float mk = tmk[i], mv = tmv[i], mr = tmr[i];
    xk[i] = xi * mk + si * (1.0f - mk);
    xv[i] = xi * mv + si * (1.0f - mv);
    xr[i] = xi * mr + si * (1.0f - mr);
  }
}

// ---------------------------------------------------------------------------
// GEMV via V_WMMA_F32_16X16X4_F32.
// One block = one 16-row output tile. blockIdx.y selects {W, x, y} triple.
// 8 waves split the K=4096 reduction; per step:
//   A (16x4 of W): lane L<16 -> row L, k 0..1 ; lane L+16 -> row L, k 2..3
//   B (4x16): x-chunk replicated into all 16 columns -> every C column equals
//   the partial dot product. Two accumulators hide the WMMA RAW chain.
// ---------------------------------------------------------------------------
__global__ __launch_bounds__(BLOCK) void gemv_wmma_kernel(
    const float* __restrict__ W0, const float* __restrict__ W1,
    const float* __restrict__ W2,
    const float* __restrict__ x0, const float* __restrict__ x1,
    const float* __restrict__ x2,
    float* __restrict__ y0, float* __restrict__ y1, float* __restrict__ y2)
{
  __shared__ __align__(16) float lds_x[DSZ];
  __shared__ float lds_red[WAVES * TILE];

  const float* W; const float* xvec; float* y;
  switch (blockIdx.y) {
    case 0:  W = W0; xvec = x0; y = y0; break;
    case 1:  W = W1; xvec = x1; y = y1; break;
    default: W = W2; xvec = x2; y = y2; break;
  }

  // Stage the 16KB vector into LDS (coalesced b128).
  for (int i = threadIdx.x; i < DSZ / 4; i += BLOCK)
    ((v4f*)lds_x)[i] = ((const v4f*)xvec)[i];
  __syncthreads();

  const int lane = threadIdx.x & 31;
  const int wave = threadIdx.x >> 5;
  const int row  = (blockIdx.x << 4) + (lane & 15);
  const int koff = (lane >> 4) << 1;          // 0 for lanes 0-15, 2 for 16-31

  const float* aBase = W + (size_t)row * DSZ + koff;

  v8f c0 = {}; v8f c1 = {};
  const int kBeg = wave * KSEG;
  const int kEnd = kBeg + KSEG;
  for (int kc = kBeg; kc < kEnd; kc += 8) {
    v2f a0 = *(const v2f*)(aBase + kc);
    v2f b0 = *(const v2f*)(&lds_x[kc + koff]);
    v2f a1 = *(const v2f*)(aBase + kc + 4);
    v2f b1 = *(const v2f*)(&lds_x[kc + 4 + koff]);
    c0 = __builtin_amdgcn_wmma_f32_16x16x4_f32(false, a0, false, b0,
                                               (short)0, c0, false, false);
    c1 = __builtin_amdgcn_wmma_f32_16x16x4_f32(false, a1, false, b1,
                                               (short)0, c1, false, false);
  }
  v8f c = c0 + c1;

  // Every column of C is identical: lane 0 holds rows 0..7 (C VGPRs 0..7),
  // lane 16 holds rows 8..15. Cross-wave reduce through LDS.
  if ((lane & 15) == 0) {
    int half = lane >> 4;
#pragma unroll
    for (int r = 0; r < 8; ++r)
      lds_red[wave * TILE + half * 8 + r] = c[r];
  }
  __syncthreads();

  if (threadIdx.x < TILE) {
    float s = 0.0f;
#pragma unroll
    for (int w = 0; w < WAVES; ++w) s += lds_red[w * TILE + threadIdx.x];
    y[(blockIdx.x << 4) + threadIdx.x] = s;
  }
}

// ---------------------------------------------------------------------------
// Kernel 2: stabilized WKV elementwise math.
// Writes rkv (input to w_out GEMV) into workspace and the state outputs
// (x, new_aa, new_bb, new_pp) into d_out[D..5D).
// ---------------------------------------------------------------------------
__global__ __launch_bounds__(BLOCK) void wkv_kernel(
    const float* __restrict__ x,  const float* __restrict__ aa,
    const float* __restrict__ bb, const float* __restrict__ pp,
    const float* __restrict__ t_decay, const float* __restrict__ t_first,
    const float* __restrict__ k, const float* __restrict__ v,
    const float* __restrict__ rraw,
    float* __restrict__ rkv, float* __restrict__ out)
{
  int i = blockIdx.x * blockDim.x + threadIdx.x;
  if (i >= DSZ) return;

  float ki = k[i], vi = v[i];
  float aai = aa[i], bbi = bb[i], ppi = pp[i];
  float r = 1.0f / (1.0f + expf(-rraw[i]));

  // current-token output path (bonus t_first)
  float ww = t_first[i] + ki;
  float p  = fmaxf(ppi, ww);
  float e1 = expf(ppi - p);
  float e2 = expf(ww - p);
  float a  = e1 * aai + e2 * vi;
  float b  = e1 * bbi + e2;

  // state update path (decay)
  float ww2 = ppi + t_decay[i];
  float p2  = fmaxf(ww2, ki);
  float e1b = expf(ww2 - p2);
  float e2b = expf(ki - p2);

  rkv[i] = r * (a / b);

  out[1 * DSZ + i] = x[i];                  // new state_xx
  out[2 * DSZ + i] = e1b * aai + e2b * vi;  // new aa
  out[3 * DSZ + i] = e1b * bbi + e2b;       // new bb
  out[4 * DSZ + i] = p2;                    // new pp
}

// ---------------------------------------------------------------------------
extern "C" void kernel_launch(void* const* d_in, const int* in_sizes, int n_in,
                              void* d_out, int out_size, void* d_ws,
                              size_t ws_size, hipStream_t stream)
{
  const float* x       = (const float*)d_in[0];
  const float* sxx     = (const float*)d_in[1];
  const float* aa      = (const float*)d_in[2];
  const float* bb      = (const float*)d_in[3];
  const float* pp      = (const float*)d_in[4];
  const float* w_key   = (const float*)d_in[5];
  const float* w_val   = (const float*)d_in[6];
  const float* w_rec   = (const float*)d_in[7];
  const float* w_out   = (const float*)d_in[8];
  const float* t_decay = (const float*)d_in[9];
  const float* t_first = (const float*)d_in[10];
  const float* t_mix_k = (const float*)d_in[11];
  const float* t_mix_v = (const float*)d_in[12];
  const float* t_mix_r = (const float*)d_in[13];

  float* out = (float*)d_out;
  float* ws  = (float*)d_ws;
  float* xk  = ws + 0 * DSZ;
  float* xv  = ws + 1 * DSZ;
  float* xr  = ws + 2 * DSZ;
  float* kk  = ws + 3 * DSZ;
  float* vv  = ws + 4 * DSZ;
  float* rr  = ws + 5 * DSZ;
  float* rkv = ws + 6 * DSZ;

  // 1) token mixing
  mix_kernel<<<DSZ / BLOCK, BLOCK, 0, stream>>>(x, sxx, t_mix_k, t_mix_v,
                                                t_mix_r, xk, xv, xr);

  // 2) three independent GEMVs (k, v, r_raw)
  dim3 g1(DSZ / TILE, 3);
  gemv_wmma_kernel<<<g1, BLOCK, 0, stream>>>(w_key, w_val, w_rec,
                                             xk, xv, xr,
                                             kk, vv, rr);

  // 3) WKV elementwise (state outputs + rkv)
  wkv_kernel<<<DSZ / BLOCK, BLOCK, 0, stream>>>(x, aa, bb, pp, t_decay,
                                                t_first, kk, vv, rr,
                                                rkv, out);

  // 4) y = w_out @ rkv  -> d_out[0..D)
  dim3 g2(DSZ / TILE, 1);
  gemv_wmma_kernel<<<g2, BLOCK, 0, stream>>>(w_out, w_out, w_out,
                                             rkv, rkv, rkv,
                                             out, out, out);
}